// GNNModel_20701742366745
// MI455X (gfx1250) — compile-verified
//
#include <hip/hip_runtime.h>
#include <math.h>

typedef __attribute__((ext_vector_type(2))) float v2f;
typedef __attribute__((ext_vector_type(8))) float v8f;
typedef __attribute__((ext_vector_type(4))) unsigned int v4u;
typedef __attribute__((ext_vector_type(8))) int v8i;
typedef __attribute__((ext_vector_type(4))) int v4i;

static constexpr int BATCH = 32;
static constexpr int NNODE = 1024;
static constexpr int DIM   = 128;           // D == H == 128
static constexpr int KNN   = 5;
static constexpr int ROWS  = BATCH * NNODE; // 32768
static constexpr int PROW  = 132;           // padded LDS row stride (floats)

__device__ __forceinline__ v8f wmma4(v2f a, v2f b, v8f c) {
  // V_WMMA_F32_16X16X4_F32 : D = A(16x4) * B(4x16) + C(16x16), all fp32
  return __builtin_amdgcn_wmma_f32_16x16x4_f32(false, a, false, b, (short)0, c,
                                               false, false);
}

// Generic->LDS offset: shared-aperture flat addr keeps the LDS byte offset in
// its low 32 bits (CDNA5 flat addressing, aperture in addr[63:32]).
__device__ __forceinline__ unsigned lds_off(const void* p) {
  return (unsigned)(size_t)p;
}

// TDM: DMA one 16-row x 128-col fp32 panel (row-major, row stride 128) from
// global memory into LDS at byte offset `dst`, padding +4 DWORDs per 128 so
// the LDS row stride becomes PROW=132 floats (bank-conflict-free fragments).
// D# per CDNA5 ISA ch.8; amdgpu-toolchain (clang-23) 6-arg builtin:
//   (uint32x4 g0, int32x8 g1, int32x4 g2, int32x4 g3, int32x8, i32 cpol)
__device__ __forceinline__ void tdm_load_panel(const float* src, unsigned dst) {
  const unsigned long long ga = (unsigned long long)(size_t)src;
  v4u g0;
  g0[0] = 1u;                                  // count=1, user mode
  g0[1] = dst;                                 // lds_addr (bytes)
  g0[2] = (unsigned)(ga & 0xffffffffu);        // global_addr[31:0]
  g0[3] = (unsigned)((ga >> 32) & 0x01ffffffu) // global_addr[56:32]
          | (2u << 30);                        // type = 2 ("image")
  v8i g1;
  g1[0] = (int)((2u << 16)        // data_size = 4 bytes
              | (1u << 20)        // pad_enable
              | (6u << 22)        // pad_interval: 128 DWORDs
              | (3u << 25));      // pad_amount : 4 DWORDs
  g1[1] = (int)(128u << 16);      // tensor_dim0 = 128      (bits 79:48 lo)
  g1[2] = (int)(1024u << 16);     // tensor_dim1 = 1024     (bits 111:80 lo)
  g1[3] = (int)(128u << 16);      // tensor_dim1 hi=0 | tile_dim0 = 128
  g1[4] = 16;                     // tile_dim1 = 16 | tile_dim2 = 0
  g1[5] = 128;                    // tensor_dim0_stride = 128 (lo 32)
  g1[6] = 0;                      // stride0 hi | stride1 lo (unused, 2D tile)
  g1[7] = 0;
  const v4i z4 = {0, 0, 0, 0};            // groups 2/3 unused (<=2D tensor)
  const v8i z8 = {0, 0, 0, 0, 0, 0, 0, 0}; // extra group (unused, zero)
  __builtin_amdgcn_tensor_load_to_lds(g0, g1, z4, z4, z8, 0);
}

// Sorted top-5 insert on scalar registers.
#define TOP5_INSERT(v, m)                                                     \
  if ((v) > tv4) {                                                            \
    if ((v) > tv2) {                                                          \
      if ((v) > tv1) {                                                        \
        if ((v) > tv0) { tv4=tv3;ti4=ti3; tv3=tv2;ti3=ti2; tv2=tv1;ti2=ti1;   \
                         tv1=tv0;ti1=ti0; tv0=(v);ti0=(m); }                  \
        else           { tv4=tv3;ti4=ti3; tv3=tv2;ti3=ti2; tv2=tv1;ti2=ti1;   \
                         tv1=(v);ti1=(m); }                                   \
      } else           { tv4=tv3;ti4=ti3; tv3=tv2;ti3=ti2; tv2=(v);ti2=(m); } \
    } else {                                                                  \
      if ((v) > tv3)   { tv4=tv3;ti4=ti3; tv3=(v);ti3=(m); }                  \
      else             { tv4=(v);ti4=(m); }                                   \
    }                                                                         \
  }

// ---------------- Kernel 1: L2-normalize each row of x ----------------
__global__ void k_normalize(const float* __restrict__ x, float* __restrict__ xn) {
  const int wave = (blockIdx.x * blockDim.x + threadIdx.x) >> 5;
  const int lane = threadIdx.x & 31;
  float4 v = ((const float4*)(x + (size_t)wave * DIM))[lane];
  float ss = v.x * v.x + v.y * v.y + v.z * v.z + v.w * v.w;
#pragma unroll
  for (int off = 16; off > 0; off >>= 1) ss += __shfl_xor(ss, off, 32);
  const float inv = 1.0f / fmaxf(sqrtf(ss), 1e-12f);
  float4 o;
  o.x = v.x * inv; o.y = v.y * inv; o.z = v.z * inv; o.w = v.w * inv;
  ((float4*)(xn + (size_t)wave * DIM))[lane] = o;
}

// ---------------- Kernel 2: fused cosine-sim GEMM + top-5 ----------------
// 2-wave block owns a 16-row tile of sim[b]; waves split the 64 column tiles
// (ct = w, w+2, ...). B panels are staged global->LDS with double-buffered
// TDM (tensor_load_to_lds / s_wait_tensorcnt), fragments fed to a 32-deep
// v_wmma_f32_16x16x4_f32 chain from LDS. Per-row top-5 kept in registers.
__global__ void __launch_bounds__(64)
k_knn(const float* __restrict__ xn, int* __restrict__ knn_idx) {
  __shared__ float panel[2][2][16 * PROW]; // [wave][buf]: padded B panels
  __shared__ float stile[2][16][17];       // per-wave sim tile (+1 pad)
  __shared__ float mval[16][10];           // merge: 2 waves x 5 candidates
  __shared__ int   midx[16][10];
  const int w       = threadIdx.x >> 5; // 0..1
  const int lane    = threadIdx.x & 31;
  const int wid     = blockIdx.x;       // row-tile id, BATCH*64 total
  const int b       = wid >> 6;
  const int rowbase = (wid & 63) * 16;
  const int lm      = lane & 15;
  const int koff    = (lane < 16) ? 0 : 2; // 16x4 f32 fragment layout
  const float* xb   = xn + (size_t)b * NNODE * DIM;

  // Resident A panel: 16 rows x 128 K as 32 (16x4) fragments (one-time).
  v2f a[32];
#pragma unroll
  for (int c = 0; c < 32; ++c)
    a[c] = *(const v2f*)(xb + (size_t)(rowbase + lm) * DIM + 4 * c + koff);

  const unsigned pbase[2] = { lds_off(&panel[w][0][0]), lds_off(&panel[w][1][0]) };
  // Prologue: TDM-load this wave's first column panel (ct = w).
  tdm_load_panel(xb + (size_t)(w * 16) * DIM, pbase[0]);

  float tv0 = -2.0f, tv1 = -2.0f, tv2 = -2.0f, tv3 = -2.0f, tv4 = -2.0f;
  int   ti0 = 0, ti1 = 0, ti2 = 0, ti3 = 0, ti4 = 0;
  const int gi = rowbase + lm;

  int cur = 0;
  for (int t = 0; t < 32; ++t) {
    const int ct      = w + 2 * t;
    const int colbase = ct * 16;
    // Issue next panel's TDM, then wait for the current one (TDM in-order).
    if (t < 31) {
      tdm_load_panel(xb + (size_t)(colbase + 32) * DIM, pbase[cur ^ 1]);
      __builtin_amdgcn_s_wait_tensorcnt(1);
    } else {
      __builtin_amdgcn_s_wait_tensorcnt(0);
    }

    // Phase 1: burst ds_load of 32 B fragments (padded stride -> no conflicts)
    const float* pb = &panel[w][cur][0];
    v2f bv[32];
#pragma unroll
    for (int c = 0; c < 32; ++c)
      bv[c] = *(const v2f*)(pb + lm * PROW + 4 * c + koff);
    __builtin_amdgcn_sched_barrier(0); // keep loads ahead of the WMMA chain

    // Phase 2: back-to-back fp32 WMMA accumulation chain.
    v8f acc = {};
#pragma unroll
    for (int c = 0; c < 32; ++c) acc = wmma4(a[c], bv[c], acc);
    __builtin_amdgcn_sched_barrier(0);

    // Spill C tile: VGPR r -> row r / r+8 (lane half), col = lane%16.
    const int rofs = (lane >> 4) << 3;
#pragma unroll
    for (int r = 0; r < 8; ++r) stile[w][r + rofs][lm] = acc[r];
    asm volatile("s_wait_dscnt 0" ::: "memory");

    if (lane < 16) { // divergent scan; reconverges before next WMMA
#pragma unroll 4
      for (int j = 0; j < 16; ++j) {
        const int   m = colbase + j;
        const float v = stile[w][lane][j];
        if (m != gi) { TOP5_INSERT(v, m) }
      }
    }
    cur ^= 1;
  }

  // Publish per-wave candidates; merge 10 -> 5 per row.
  if (lane < 16) {
    mval[lane][w*5+0]=tv0; midx[lane][w*5+0]=ti0;
    mval[lane][w*5+1]=tv1; midx[lane][w*5+1]=ti1;
    mval[lane][w*5+2]=tv2; midx[lane][w*5+2]=ti2;
    mval[lane][w*5+3]=tv3; midx[lane][w*5+3]=ti3;
    mval[lane][w*5+4]=tv4; midx[lane][w*5+4]=ti4;
  }
  __syncthreads();
  if (threadIdx.x < 16) {
    float tv0=-2.0f,tv1=-2.0f,tv2=-2.0f,tv3=-2.0f,tv4=-2.0f;
    int   ti0=0,ti1=0,ti2=0,ti3=0,ti4=0;
#pragma unroll
    for (int q = 0; q < 10; ++q) {
      const float v = mval[threadIdx.x][q];
      const int   m = midx[threadIdx.x][q];
      TOP5_INSERT(v, m)
    }
    const size_t o = (size_t)(b * NNODE + rowbase + threadIdx.x) * KNN;
    knn_idx[o + 0] = ti0; knn_idx[o + 1] = ti1; knn_idx[o + 2] = ti2;
    knn_idx[o + 3] = ti3; knn_idx[o + 4] = ti4;
  }
}

// ---------------- Kernel 3: degree-normalized aggregation (deg == 6) --------
__global__ void k_aggregate(const float* __restrict__ f,
                            const int* __restrict__ knn_idx,
                            float* __restrict__ agg) {
  const int node = (blockIdx.x * blockDim.x + threadIdx.x) >> 5;
  const int lane = threadIdx.x & 31;
  const int b    = node >> 10; // N = 1024
  const float* fb = f + ((size_t)b * NNODE) * DIM;
  float4 acc = ((const float4*)(f + (size_t)node * DIM))[lane]; // self loop
#pragma unroll
  for (int q = 0; q < KNN; ++q) {
    const int j = knn_idx[(size_t)node * KNN + q];
    float4 nv = ((const float4*)(fb + (size_t)j * DIM))[lane];
    acc.x += nv.x; acc.y += nv.y; acc.z += nv.z; acc.w += nv.w;
  }
  const float s = 1.0f / 6.0f;
  acc.x *= s; acc.y *= s; acc.z *= s; acc.w *= s;
  ((float4*)(agg + (size_t)node * DIM))[lane] = acc;
}

// ---------------- Kernel 4: out = resid + relu(agg @ W^T + bias) ------------
__global__ void __launch_bounds__(128)
k_dense(const float* __restrict__ agg, const float* __restrict__ W,
        const float* __restrict__ bias, const float* __restrict__ resid,
        float* __restrict__ out) {
  const int lane    = threadIdx.x & 31;
  const int wid     = blockIdx.x * 4 + (threadIdx.x >> 5);
  const int rowbase = (wid >> 3) * 16; // over B*N rows
  const int hbase   = (wid & 7) * 16;  // over H=128 cols
  const int lm      = lane & 15;
  const int koff    = (lane < 16) ? 0 : 2;

  const float* arow = agg + (size_t)(rowbase + lm) * DIM + koff;
  const float* wrow = W + (size_t)(hbase + lm) * DIM + koff; // B(k,h)=W[h][k]
  v2f av[32], bv[32];
#pragma unroll
  for (int c = 0; c < 32; ++c) av[c] = *(const v2f*)(arow + 4 * c);
#pragma unroll
  for (int c = 0; c < 32; ++c) bv[c] = *(const v2f*)(wrow + 4 * c);
  __builtin_amdgcn_sched_barrier(0); // force the 64-load burst before WMMA
  v8f acc = {};
#pragma unroll
  for (int c = 0; c < 32; ++c) acc = wmma4(av[c], bv[c], acc);
  __builtin_amdgcn_sched_barrier(0);

  const float bc   = bias[hbase + lm];
  const int   rofs = (lane >> 4) << 3;
#pragma unroll
  for (int r = 0; r < 8; ++r) {
    const size_t o = (size_t)(rowbase + r + rofs) * DIM + hbase + lm;
    out[o] = fmaxf(acc[r] + bc, 0.0f) + resid[o];
  }
}

extern "C" void kernel_launch(void* const* d_in, const int* in_sizes, int n_in,
                              void* d_out, int out_size, void* d_ws, size_t ws_size,
                              hipStream_t stream) {
  (void)in_sizes; (void)n_in; (void)out_size; (void)ws_size;
  const float* x  = (const float*)d_in[0];
  const float* W1 = (const float*)d_in[1];
  const float* b1 = (const float*)d_in[2];
  const float* W2 = (const float*)d_in[3];
  const float* b2 = (const float*)d_in[4];
  float* out = (float*)d_out;

  const size_t feat = (size_t)ROWS * DIM;
  float* xn  = (float*)d_ws;      // 16.8 MB
  float* agg = xn + feat;         // 16.8 MB
  float* h1  = agg + feat;        // 16.8 MB
  int*   knn = (int*)(h1 + feat); // 0.66 MB

  k_normalize<<<ROWS / 8, 256, 0, stream>>>(x, xn);
  k_knn<<<BATCH * 64, 64, 0, stream>>>(xn, knn);

  k_aggregate<<<ROWS / 8, 256, 0, stream>>>(x, knn, agg);
  k_dense<<<(ROWS / 16 * 8) / 4, 128, 0, stream>>>(agg, W1, b1, x, h1);

  k_aggregate<<<ROWS / 8, 256, 0, stream>>>(h1, knn, agg);
  k_dense<<<(ROWS / 16 * 8) / 4, 128, 0, stream>>>(agg, W2, b2, h1, out);
}